// GConv_E_57475252355659
// MI455X (gfx1250) — compile-verified
//
#include <hip/hip_runtime.h>
#include <stddef.h>

// ---------------- problem constants (match reference) ----------------
#define N_NODESC   100000
#define N_EDGESC   1280000
#define HDIM       64
#define H2DIM      128
#define NLAYERS    5
#define NGRAPHS    512
#define ATOM_COLS  9
#define ATOM_VOCAB 128
#define BOND_COLS  3
#define BOND_VOCAB 8
#define BN_EPSF    1e-5f

typedef float v2f __attribute__((ext_vector_type(2)));
typedef float v8f __attribute__((ext_vector_type(8)));

__device__ __forceinline__ void atomicAddF32(float* p, float v) {
  // lowers to global_atomic_add_f32 on gfx1250 (relaxed, device scope)
  __hip_atomic_fetch_add(p, v, __ATOMIC_RELAXED, __HIP_MEMORY_SCOPE_AGENT);
}

// ---------------- utility: zero a float buffer ----------------
__global__ void zero_f32(float* __restrict__ p, int n) {
  int i = blockIdx.x * blockDim.x + threadIdx.x;
  if (i < n) p[i] = 0.0f;
}

// ---------------- atom encoder: z[n,h] = sum_c atom_emb[c, x[n,c], h] ----------------
// one wave per node; lane covers h = lane and h = lane+32
__global__ void __launch_bounds__(256) atom_encode(const int* __restrict__ x,
                                                   const float* __restrict__ emb,
                                                   float* __restrict__ z) {
  int node = (blockIdx.x * blockDim.x + threadIdx.x) >> 5;
  int lane = threadIdx.x & 31;
  if (node >= N_NODESC) return;
  float a0 = 0.0f, a1 = 0.0f;
#pragma unroll
  for (int c = 0; c < ATOM_COLS; ++c) {
    int v = x[node * ATOM_COLS + c];
    const float* row = emb + ((size_t)(c * ATOM_VOCAB + v)) * HDIM;
    a0 += row[lane];
    a1 += row[lane + 32];
  }
  z[(size_t)node * HDIM + lane]      = a0;
  z[(size_t)node * HDIM + lane + 32] = a1;
}

// ---------------- edge messages: aggr[dst] += relu(z[src] + e) ----------------
// bond table is 6KB -> always cache-hot; recompute e inline, never materialize it.
__global__ void __launch_bounds__(256) edge_message(const int* __restrict__ ei,
                                                    const int* __restrict__ ea,
                                                    const float* __restrict__ bemb,
                                                    const float* __restrict__ z,
                                                    float* __restrict__ aggr) {
  int e = (blockIdx.x * blockDim.x + threadIdx.x) >> 5;
  int lane = threadIdx.x & 31;
  if (e >= N_EDGESC) return;
  int src = ei[e];
  int dst = ei[N_EDGESC + e];
  int b0 = ea[e * BOND_COLS + 0];
  int b1 = ea[e * BOND_COLS + 1];
  int b2 = ea[e * BOND_COLS + 2];
  const float* r0 = bemb + (size_t)(0 * BOND_VOCAB + b0) * HDIM;
  const float* r1 = bemb + (size_t)(1 * BOND_VOCAB + b1) * HDIM;
  const float* r2 = bemb + (size_t)(2 * BOND_VOCAB + b2) * HDIM;
  float e0 = r0[lane] + r1[lane] + r2[lane];
  float e1 = r0[lane + 32] + r1[lane + 32] + r2[lane + 32];
  float m0 = z[(size_t)src * HDIM + lane] + e0;
  float m1 = z[(size_t)src * HDIM + lane + 32] + e1;
  m0 = m0 > 0.0f ? m0 : 0.0f;
  m1 = m1 > 0.0f ? m1 : 0.0f;
  atomicAddF32(&aggr[(size_t)dst * HDIM + lane], m0);
  atomicAddF32(&aggr[(size_t)dst * HDIM + lane + 32], m1);
}

// ---------------- GEMM1: h1 = (z + aggr) @ W1 + b1 ; [N,64]x[64,128] ----------------
// fp32 WMMA 16x16x4. One wave per 16x16 output tile; block = 8 waves = one M-tile row.
// A layout (ISA 7.12.2, 32-bit A 16x4): lanes 0-15 hold M=lane {K=0 in v0, K=1 in v1},
// lanes 16-31 hold M=lane-16 {K=2 in v0, K=3 in v1}. B mirrors with N across lanes.
__global__ void __launch_bounds__(256) gemm_zaggr_w1(const float* __restrict__ z,
                                                     const float* __restrict__ aggr,
                                                     const float* __restrict__ W,
                                                     const float* __restrict__ bias,
                                                     float* __restrict__ out) {
  int mtile = blockIdx.x;                 // 6250 tiles of 16 rows
  int ntile = threadIdx.x >> 5;           // 8 waves -> 8 column tiles of 16
  int lane  = threadIdx.x & 31;
  int row   = mtile * 16 + (lane & 15);
  int khalf = (lane >> 4) * 2;            // 0 for lanes 0-15, 2 for lanes 16-31
  int col   = ntile * 16 + (lane & 15);
  v8f acc = {};
#pragma unroll
  for (int k0 = 0; k0 < HDIM; k0 += 4) {
    int ka = k0 + khalf;
    v2f a, b;
    a.x = z[(size_t)row * HDIM + ka]     + aggr[(size_t)row * HDIM + ka];
    a.y = z[(size_t)row * HDIM + ka + 1] + aggr[(size_t)row * HDIM + ka + 1];
    b.x = W[(size_t)ka * H2DIM + col];
    b.y = W[(size_t)(ka + 1) * H2DIM + col];
    acc = __builtin_amdgcn_wmma_f32_16x16x4_f32(false, a, false, b,
                                                (short)0, acc, false, false);
  }
  float bv = bias[col];
  int rbase = mtile * 16 + ((lane >> 4) * 8);   // D: vgpr j -> M=j (+8 for hi lanes)
#pragma unroll
  for (int j = 0; j < 8; ++j)
    out[(size_t)(rbase + j) * H2DIM + col] = acc[j] + bv;
}

// ---------------- BN stats: per-channel sum / sumsq ----------------
__global__ void __launch_bounds__(256) bn_stats(const float* __restrict__ h, int C,
                                                float* __restrict__ sum,
                                                float* __restrict__ sq) {
  int tpr  = 256 / C;                       // rows covered per block step (C divides 256)
  int c    = threadIdx.x & (C - 1);
  int r    = blockIdx.x * tpr + threadIdx.x / C;
  int step = gridDim.x * tpr;
  float s = 0.0f, q = 0.0f;
  for (; r < N_NODESC; r += step) {
    float v = h[(size_t)r * C + c];
    s += v;
    q += v * v;
  }
  atomicAddF32(&sum[c], s);
  atomicAddF32(&sq[c], q);
}

// ---------------- fold BN into scale/shift: y = h*sc + sh ----------------
__global__ void bn_coeffs(const float* __restrict__ sum, const float* __restrict__ sq,
                          const float* __restrict__ gamma, const float* __restrict__ beta,
                          int C, float invN, float* __restrict__ sc, float* __restrict__ sh) {
  int c = threadIdx.x;
  if (c < C) {
    float mu  = sum[c] * invN;
    float var = sq[c] * invN - mu * mu;     // biased variance, as reference
    float rs  = rsqrtf(var + BN_EPSF);
    float s   = rs * gamma[c];
    sc[c] = s;
    sh[c] = beta[c] - mu * s;
  }
}

// ---------------- GEMM2: h2 = relu(BN(h1)) @ W2 + b2 ; [N,128]x[128,64] ----------------
// BN1 + ReLU fused into the A-matrix load.
__global__ void __launch_bounds__(128) gemm_h_w2(const float* __restrict__ h1,
                                                 const float* __restrict__ sc,
                                                 const float* __restrict__ sh,
                                                 const float* __restrict__ W,
                                                 const float* __restrict__ bias,
                                                 float* __restrict__ out) {
  int mtile = blockIdx.x;                 // 6250
  int ntile = threadIdx.x >> 5;           // 4 waves -> 4 column tiles of 16
  int lane  = threadIdx.x & 31;
  int row   = mtile * 16 + (lane & 15);
  int khalf = (lane >> 4) * 2;
  int col   = ntile * 16 + (lane & 15);
  v8f acc = {};
#pragma unroll
  for (int k0 = 0; k0 < H2DIM; k0 += 4) {
    int ka = k0 + khalf;
    float a0 = h1[(size_t)row * H2DIM + ka]     * sc[ka]     + sh[ka];
    float a1 = h1[(size_t)row * H2DIM + ka + 1] * sc[ka + 1] + sh[ka + 1];
    v2f a, b;
    a.x = a0 > 0.0f ? a0 : 0.0f;
    a.y = a1 > 0.0f ? a1 : 0.0f;
    b.x = W[(size_t)ka * HDIM + col];
    b.y = W[(size_t)(ka + 1) * HDIM + col];
    acc = __builtin_amdgcn_wmma_f32_16x16x4_f32(false, a, false, b,
                                                (short)0, acc, false, false);
  }
  float bv = bias[col];
  int rbase = mtile * 16 + ((lane >> 4) * 8);
#pragma unroll
  for (int j = 0; j < 8; ++j)
    out[(size_t)(rbase + j) * HDIM + col] = acc[j] + bv;
}

// ---------------- finalize: z = BN(h2) (+relu), write z_cat slice, pool g_cat ----------------
__global__ void __launch_bounds__(256) finalize(const float* __restrict__ h2,
                                                const float* __restrict__ sc,
                                                const float* __restrict__ sh,
                                                const int* __restrict__ batch,
                                                int layer, int do_relu,
                                                float* __restrict__ z,
                                                float* __restrict__ out) {
  int node = (blockIdx.x * blockDim.x + threadIdx.x) >> 5;
  int lane = threadIdx.x & 31;
  if (node >= N_NODESC) return;
  int g = batch[node];
  const int CH = NLAYERS * HDIM;  // 320
#pragma unroll
  for (int p = 0; p < 2; ++p) {
    int h = lane + p * 32;
    float v = h2[(size_t)node * HDIM + h] * sc[h] + sh[h];
    if (do_relu) v = v > 0.0f ? v : 0.0f;
    z[(size_t)node * HDIM + h] = v;
    out[(size_t)node * CH + layer * HDIM + h] = v;
    atomicAddF32(&out[(size_t)N_NODESC * CH + (size_t)g * CH + layer * HDIM + h], v);
  }
}

// ---------------- launch ----------------
extern "C" void kernel_launch(void* const* d_in, const int* in_sizes, int n_in,
                              void* d_out, int out_size, void* d_ws, size_t ws_size,
                              hipStream_t stream) {
  const int*   x        = (const int*)d_in[0];
  const int*   ei       = (const int*)d_in[1];
  const int*   ea       = (const int*)d_in[2];
  const int*   batch    = (const int*)d_in[3];
  const float* atom_emb = (const float*)d_in[4];
  const float* bond_emb = (const float*)d_in[5];
  const float* W1       = (const float*)d_in[6];
  const float* b1       = (const float*)d_in[7];
  const float* g1       = (const float*)d_in[8];
  const float* be1      = (const float*)d_in[9];
  const float* W2       = (const float*)d_in[10];
  const float* b2       = (const float*)d_in[11];
  const float* gbn      = (const float*)d_in[12];
  const float* bbn      = (const float*)d_in[13];
  float* out = (float*)d_out;

  // workspace layout (floats)
  float* z    = (float*)d_ws;                          // N*64
  float* aggr = z    + (size_t)N_NODESC * HDIM;        // N*64
  float* h1   = aggr + (size_t)N_NODESC * HDIM;        // N*128
  float* h2   = h1   + (size_t)N_NODESC * H2DIM;       // N*64
  float* st   = h2   + (size_t)N_NODESC * HDIM;        // 768 stat floats
  float* sum1 = st;        float* sq1 = st + 128;
  float* sc1  = st + 256;  float* sh1 = st + 384;
  float* sum2 = st + 512;  float* sq2 = st + 576;
  float* sc2  = st + 640;  float* sh2 = st + 704;

  const float invN = 1.0f / (float)N_NODESC;
  const int gcatN  = NGRAPHS * NLAYERS * HDIM;          // 512*320
  const int aggrN  = N_NODESC * HDIM;

  // g_cat region is pooled with atomics -> must start at zero every call
  zero_f32<<<(gcatN + 255) / 256, 256, 0, stream>>>(out + (size_t)N_NODESC * NLAYERS * HDIM, gcatN);

  // node features
  atom_encode<<<(N_NODESC * 32) / 256, 256, 0, stream>>>(x, atom_emb, z);

  for (int l = 0; l < NLAYERS; ++l) {
    // zero aggregation buffer + BN stats
    zero_f32<<<(aggrN + 255) / 256, 256, 0, stream>>>(aggr, aggrN);
    zero_f32<<<3, 256, 0, stream>>>(st, 768);

    // message passing (wave per edge; exact, no tail)
    edge_message<<<(N_EDGESC * 32) / 256, 256, 0, stream>>>(ei, ea, bond_emb, z, aggr);

    // MLP part 1 (fp32 WMMA), then BN1 stats/coeffs
    gemm_zaggr_w1<<<N_NODESC / 16, 256, 0, stream>>>(
        z, aggr, W1 + (size_t)l * HDIM * H2DIM, b1 + (size_t)l * H2DIM, h1);
    bn_stats<<<1024, 256, 0, stream>>>(h1, H2DIM, sum1, sq1);
    bn_coeffs<<<1, 128, 0, stream>>>(sum1, sq1, g1 + (size_t)l * H2DIM, be1 + (size_t)l * H2DIM,
                                     H2DIM, invN, sc1, sh1);

    // MLP part 2 with fused BN1+ReLU on A-load (fp32 WMMA), then outer BN stats/coeffs
    gemm_h_w2<<<N_NODESC / 16, 128, 0, stream>>>(
        h1, sc1, sh1, W2 + (size_t)l * H2DIM * HDIM, b2 + (size_t)l * HDIM, h2);
    bn_stats<<<1024, 256, 0, stream>>>(h2, HDIM, sum2, sq2);
    bn_coeffs<<<1, 64, 0, stream>>>(sum2, sq2, gbn + (size_t)l * HDIM, bbn + (size_t)l * HDIM,
                                    HDIM, invN, sc2, sh2);

    // z update + z_cat slice + g_cat pooling
    finalize<<<(N_NODESC * 32) / 256, 256, 0, stream>>>(
        h2, sc2, sh2, batch, l, (l != NLAYERS - 1) ? 1 : 0, z, out);
  }
}